// O3TensorProduct_81844896793185
// MI455X (gfx1250) — compile-verified
//
#include <hip/hip_runtime.h>

typedef float v2f __attribute__((ext_vector_type(2)));
typedef float v8f __attribute__((ext_vector_type(8)));

#define M0        128
#define M1        64
#define ROWSTRIDE 320            // M0 + 3*M1
#define NKK       48             // stacked K (192) / 4
#define INV_SQRT3 0.57735026918962576f

// LDS (floats): packed-B for Ws = [W00 ; INV_SQRT3*W11] (192x128)  -> 48*8*32 pairs
//               packed-B for Wv = [W01 ; W10]            (192x64)  -> 48*4*32 pairs
#define WS_PAIRS (NKK * 8 * 32)            // 12288 pairs = 24576 floats
#define WV_PAIRS (NKK * 4 * 32)            // 6144 pairs  = 12288 floats
#define WV_OFF   (WS_PAIRS * 2)            // float offset 24576
#define LDS_FLOATS ((WS_PAIRS + WV_PAIRS) * 2)  // 36864 floats = 147456 B

// One group's GEMM: D(16xNCT*16) += A(16x192) * B(192xNCT*16), kk-outer / ct-inner.
template <int NCT, bool IS_G0>
__device__ __forceinline__ void run_group(
    const float* __restrict__ row1, const v2f* __restrict__ pk,
    float fac, float s2, float v2x, float v2y, float v2z, int kc,
    int half, v8f (&c)[NCT])
{
    const v8f zero = {};
    #pragma unroll
    for (int ct = 0; ct < NCT; ++ct) c[ct] = zero;

    // K rows 0..127: fac * s1  (one b64 global load per K-step)
    #pragma unroll 4
    for (int kk = 0; kk < 32; ++kk) {
        v2f a = *(const v2f*)(row1 + 4 * kk + 2 * half) * fac;
        #pragma unroll
        for (int ct = 0; ct < NCT; ++ct) {
            v2f b = pk[(kk * NCT + ct) * 32];   // ds_load_b64, immediate-offset friendly
            c[ct] = __builtin_amdgcn_wmma_f32_16x16x4_f32(
                        false, a, false, b, (short)0, c[ct], false, false);
        }
    }
    // K rows 128..191
    #pragma unroll 4
    for (int kk = 32; kk < NKK; ++kk) {
        const int u0 = 4 * (kk - 32) + 2 * half;   // even -> 8B aligned v2f loads
        v2f a;
        if (IS_G0) {
            const float* p = row1 + M0 + 3 * u0;   // 6 contiguous floats
            v2f q0 = *(const v2f*)(p);
            v2f q1 = *(const v2f*)(p + 2);
            v2f q2 = *(const v2f*)(p + 4);
            a.x = fmaf(q0.x, v2x, fmaf(q0.y, v2y, q1.x * v2z));
            a.y = fmaf(q1.y, v2x, fmaf(q2.x, v2y, q2.y * v2z));
        } else {
            a.x = row1[M0 + 3 * u0 + kc] * s2;
            a.y = row1[M0 + 3 * (u0 + 1) + kc] * s2;
        }
        #pragma unroll
        for (int ct = 0; ct < NCT; ++ct) {
            v2f b = pk[(kk * NCT + ct) * 32];
            c[ct] = __builtin_amdgcn_wmma_f32_16x16x4_f32(
                        false, a, false, b, (short)0, c[ct], false, false);
        }
    }
}

__global__ __launch_bounds__(256) void o3tp_wmma_kernel(
    const float* __restrict__ din1, const float* __restrict__ din2,
    const float* __restrict__ W00,  const float* __restrict__ W01,
    const float* __restrict__ W10,  const float* __restrict__ W11,
    const float* __restrict__ bias, float* __restrict__ out, int N)
{
    __shared__ float lds[LDS_FLOATS];
    const int tid = threadIdx.x;

    // ---- Stage weights pre-swizzled into B-fragment order ----
    // pair p -> kk, ct, lane; value j (0/1) = Wrow(4kk + 2*(lane>>4) + j)[16ct + (lane&15)]
    for (int p = tid; p < WS_PAIRS; p += 256) {
        const int kk = p >> 8, rem = p & 255;
        const int ct = rem >> 5, ln = rem & 31;
        const int K0 = 4 * kk + 2 * (ln >> 4);
        const int col = 16 * ct + (ln & 15);
        const float v0 = (K0 < M0) ? W00[K0 * M0 + col]
                                   : INV_SQRT3 * W11[(K0 - M0) * M0 + col];
        const float v1 = (K0 + 1 < M0) ? W00[(K0 + 1) * M0 + col]
                                       : INV_SQRT3 * W11[(K0 + 1 - M0) * M0 + col];
        lds[2 * p]     = v0;
        lds[2 * p + 1] = v1;
    }
    for (int p = tid; p < WV_PAIRS; p += 256) {
        const int kk = p >> 7, rem = p & 127;
        const int ct = rem >> 5, ln = rem & 31;
        const int K0 = 4 * kk + 2 * (ln >> 4);
        const int col = 16 * ct + (ln & 15);
        const float v0 = (K0 < M0) ? W01[K0 * M1 + col] : W10[(K0 - M0) * M1 + col];
        const float v1 = (K0 + 1 < M0) ? W01[(K0 + 1) * M1 + col]
                                       : W10[(K0 + 1 - M0) * M1 + col];
        lds[WV_OFF + 2 * p]     = v0;
        lds[WV_OFF + 2 * p + 1] = v1;
    }
    __syncthreads();

    const int wave = tid >> 5;
    const int lane = tid & 31;
    const int m    = lane & 15;
    const int half = lane >> 4;

    const int n0 = blockIdx.x * 128 + wave * 16;
    if (n0 >= N) return;                           // uniform per-wave

    int nRow = n0 + m; if (nRow >= N) nRow = N - 1;
    const float* row1 = din1 + (size_t)nRow * ROWSTRIDE;
    const float* row2 = din2 + (size_t)nRow * 4;
    const float s2  = row2[0];
    const float v2x = row2[1];
    const float v2y = row2[2];
    const float v2z = row2[3];

    const v2f* pkS = (const v2f*)(lds) + lane;
    const v2f* pkV = (const v2f*)(lds + WV_OFF) + lane;

    // ---- Group 0: out0 = [s2*s1 | d] @ [W00 ; k*W11] + bias ----
    {
        v8f c[8];
        run_group<8, true>(row1, pkS, s2, s2, v2x, v2y, v2z, 0, half, c);
        #pragma unroll
        for (int ct = 0; ct < 8; ++ct) {
            const int col = ct * 16 + m;
            const float bcol = bias[col];
            #pragma unroll
            for (int v = 0; v < 8; ++v) {
                const int r = n0 + v + 8 * half;
                if (r < N) out[(size_t)r * ROWSTRIDE + col] = c[ct][v] + bcol;
            }
        }
    }
    // ---- Groups 1..3: outv_k = [v2k*s1 | s2*v1k] @ [W01 ; W10] ----
    for (int g = 1; g <= 3; ++g) {
        const float fac = (g == 1) ? v2x : ((g == 2) ? v2y : v2z);
        v8f c[4];
        run_group<4, false>(row1, pkV, fac, s2, v2x, v2y, v2z, g - 1, half, c);
        #pragma unroll
        for (int ct = 0; ct < 4; ++ct) {
            const int ocol = M0 + 3 * (ct * 16 + m) + (g - 1);
            #pragma unroll
            for (int v = 0; v < 8; ++v) {
                const int r = n0 + v + 8 * half;
                if (r < N) out[(size_t)r * ROWSTRIDE + ocol] = c[ct][v];
            }
        }
    }
}

extern "C" void kernel_launch(void* const* d_in, const int* in_sizes, int n_in,
                              void* d_out, int out_size, void* d_ws, size_t ws_size,
                              hipStream_t stream) {
    const float* din1 = (const float*)d_in[0];
    const float* din2 = (const float*)d_in[1];
    const float* W00  = (const float*)d_in[2];
    const float* W01  = (const float*)d_in[3];
    const float* W10  = (const float*)d_in[4];
    const float* W11  = (const float*)d_in[5];
    const float* bias = (const float*)d_in[6];
    float* out = (float*)d_out;

    const int N = in_sizes[0] / ROWSTRIDE;
    const int blocks = (N + 127) / 128;
    o3tp_wmma_kernel<<<blocks, 256, 0, stream>>>(din1, din2, W00, W01, W10, W11,
                                                 bias, out, N);
}